// pagerank_pool_35656818491451
// MI455X (gfx1250) — compile-verified
//
#include <hip/hip_runtime.h>

#define G       3008
#define NCLS    33
#define KSTEPS  10
#define B       4096
#define NNZ     (G * 32)
#define DIN     (2 * G)
#define H1N     1024
#define H2N     128

typedef float v2f __attribute__((ext_vector_type(2)));
typedef float v8f __attribute__((ext_vector_type(8)));

// ---------------- CDNA5 async global->LDS copy (guarded) ---------------------------

#if defined(__has_builtin)
#if __has_builtin(__builtin_amdgcn_global_load_async_to_lds_b128)
#define HAVE_ASYNC_LDS 1
#endif
#endif

typedef int v4i_g __attribute__((vector_size(16)));
typedef __attribute__((address_space(1))) v4i_g* as1_v4i;
typedef __attribute__((address_space(3))) v4i_g* as3_v4i;

__device__ __forceinline__ void async_copy16(const float* gsrc, float* ldst) {
#ifdef HAVE_ASYNC_LDS
    __builtin_amdgcn_global_load_async_to_lds_b128(
        (as1_v4i)const_cast<float*>(gsrc), (as3_v4i)ldst, 0, 0);
#else
    *(float4*)ldst = *(const float4*)gsrc;
#endif
}

__device__ __forceinline__ void async_wait() {
#ifdef HAVE_ASYNC_LDS
#if __has_builtin(__builtin_amdgcn_s_wait_asynccnt)
    __builtin_amdgcn_s_wait_asynccnt(0);
#else
    asm volatile("s_wait_asynccnt 0x0" ::: "memory");
#endif
#endif
}

// ---------------- CSR construction (fwd row_ptr + deterministic transposed CSR) ----

__global__ void k_zero_ptrs(int* fwd, int* tr) {
    int i = blockIdx.x * blockDim.x + threadIdx.x;
    if (i < G + 1) { fwd[i] = 0; tr[i] = 0; }
}

__global__ void k_count(const int* __restrict__ rows, const int* __restrict__ cols,
                        int* fwd, int* tr) {
    int i = blockIdx.x * blockDim.x + threadIdx.x;
    if (i < NNZ) {
        atomicAdd(&fwd[rows[i] + 1], 1);
        atomicAdd(&tr[cols[i] + 1], 1);
    }
}

// in-place inclusive scan of an (G+1)-int array; two arrays via blockIdx
__global__ void k_scan2(int* a, int* b) {
    int* p = (blockIdx.x == 0) ? a : b;
    const int n = G + 1;
    const int chunk = (n + 255) / 256;
    __shared__ int sums[256];
    __shared__ int offs[256];
    int t = threadIdx.x;
    int beg = t * chunk;
    int end = beg + chunk; if (end > n) end = n;
    int s = 0;
    for (int i = beg; i < end; ++i) s += p[i];
    sums[t] = s;
    __syncthreads();
    if (t == 0) {
        int run = 0;
        for (int i = 0; i < 256; ++i) { offs[i] = run; run += sums[i]; }
    }
    __syncthreads();
    int run = offs[t];
    for (int i = beg; i < end; ++i) { run += p[i]; p[i] = run; }
}

__global__ void k_init_cursor(const int* __restrict__ trp, int* cur) {
    int i = blockIdx.x * blockDim.x + threadIdx.x;
    if (i < G) cur[i] = trp[i];
}

__global__ void k_scatter_tr(const int* __restrict__ rows, const int* __restrict__ cols,
                             const float* __restrict__ vals, int* cur,
                             int* tcols, float* tvals, int* tidx) {
    int i = blockIdx.x * blockDim.x + threadIdx.x;
    if (i < NNZ) {
        int c = cols[i];
        int pos = atomicAdd(&cur[c], 1);
        tcols[pos] = rows[i];
        tvals[pos] = vals[i];
        tidx[pos]  = i;
    }
}

// restore deterministic order inside each transposed row (sort by original index)
__global__ void k_sort_seg(const int* __restrict__ trp, int* tcols, float* tvals, int* tidx) {
    int c = blockIdx.x * blockDim.x + threadIdx.x;
    if (c >= G) return;
    int lo = trp[c], hi = trp[c + 1];
    for (int i = lo + 1; i < hi; ++i) {
        int key = tidx[i]; int kc = tcols[i]; float kv = tvals[i];
        int j = i - 1;
        while (j >= lo && tidx[j] > key) {
            tidx[j + 1] = tidx[j]; tcols[j + 1] = tcols[j]; tvals[j + 1] = tvals[j];
            --j;
        }
        tidx[j + 1] = key; tcols[j + 1] = kc; tvals[j + 1] = kv;
    }
}

// ---------------- embed (transpose [B,G] -> [G,B]) + ReLU affine -------------------

__global__ void k_embed_t(const float* __restrict__ x, const float* __restrict__ w,
                          const float* __restrict__ bias, float* __restrict__ out) {
    __shared__ float tile[32][33];
    int tx = threadIdx.x, ty = threadIdx.y;
    int g0 = blockIdx.x * 32, b0 = blockIdx.y * 32;
    float wv = w[0], bv = bias[0];
#pragma unroll
    for (int j = 0; j < 4; ++j) {
        int g  = g0 + tx;
        int bb = b0 + ty + j * 8;
        tile[tx][ty + j * 8] = fmaxf(wv * x[(size_t)bb * G + g] + bv, 0.f);
    }
    __syncthreads();
#pragma unroll
    for (int j = 0; j < 4; ++j) {
        int g  = g0 + ty + j * 8;
        int bb = b0 + tx;
        out[(size_t)g * B + bb] = tile[ty + j * 8][tx];
    }
}

// final residual: prop[g,b] += 0.1 * relu(w*x[b,g]+bias)   (in place)
__global__ void k_axpy_embed(const float* __restrict__ x, const float* __restrict__ w,
                             const float* __restrict__ bias, float* __restrict__ prop) {
    __shared__ float tile[32][33];
    int tx = threadIdx.x, ty = threadIdx.y;
    int g0 = blockIdx.x * 32, b0 = blockIdx.y * 32;
    float wv = w[0], bv = bias[0];
#pragma unroll
    for (int j = 0; j < 4; ++j) {
        int g  = g0 + tx;
        int bb = b0 + ty + j * 8;
        tile[tx][ty + j * 8] = fmaxf(wv * x[(size_t)bb * G + g] + bv, 0.f);
    }
    __syncthreads();
#pragma unroll
    for (int j = 0; j < 4; ++j) {
        int g  = g0 + ty + j * 8;
        int bb = b0 + tx;
        size_t o = (size_t)g * B + bb;
        prop[o] = prop[o] + 0.1f * tile[ty + j * 8][tx];
    }
}

// ---------------- SpMM: y[r,:] = 0.9 * sum_k vals[k] * x[cols[k],:] ----------------
// float4 gathers; (cols, vals) metadata staged in LDS once per block.

#define SPMM_META 128

__global__ void k_spmm(const int* __restrict__ rp, const int* __restrict__ ci,
                       const float* __restrict__ vv, const float* __restrict__ x,
                       float* __restrict__ y) {
    __shared__ int   scol[SPMM_META];
    __shared__ float sval[SPMM_META];
    const int r   = blockIdx.x;
    const int tid = threadIdx.x;
    const int c4  = (blockIdx.y * 256 + tid) * 4;       // starting float column
    const int lo = rp[r], hi = rp[r + 1];
    float ax = 0.f, ay = 0.f, az = 0.f, aw = 0.f;
    for (int base = lo; base < hi; base += SPMM_META) {
        int cnt = hi - base; if (cnt > SPMM_META) cnt = SPMM_META;
        if (tid < cnt) { scol[tid] = ci[base + tid]; sval[tid] = vv[base + tid]; }
        __syncthreads();
        for (int k = 0; k < cnt; ++k) {
            const float4 xv = *(const float4*)&x[(size_t)scol[k] * B + c4];
            const float  v  = sval[k];
            ax += v * xv.x; ay += v * xv.y; az += v * xv.z; aw += v * xv.w;
        }
        __syncthreads();
    }
    float4 o; o.x = 0.9f * ax; o.y = 0.9f * ay; o.z = 0.9f * az; o.w = 0.9f * aw;
    *(float4*)&y[(size_t)r * B + c4] = o;
}

// ---------------- WMMA GEMM layer 1: out = relu([xs^T | xt^T] @ W + b) -------------
// xs/xt stored [G,B]; A[m,k] = (k<G? xs[k, m0+m] : xt[k-G, m0+m]); W is [2G, 1024].
// LDS tiles kept in global orientation so async b128 copies need no transpose:
//   sA[k][m] (stride 68), sB[k][n] (stride 68); both 16B-aligned per chunk.

__global__ void k_gemm1(const float* __restrict__ XS, const float* __restrict__ XT,
                        const float* __restrict__ W, const float* __restrict__ bias,
                        float* __restrict__ out) {
    __shared__ __align__(16) float sA[32][68];
    __shared__ __align__(16) float sB[32][68];
    const int tid  = threadIdx.x;
    const int lane = tid & 31;
    const int wv   = tid >> 5;
    const int m0 = blockIdx.x * 64;
    const int n0 = blockIdx.y * 64;
    const int mw = (wv & 3) * 16;
    const int nw = (wv >> 2) * 32;
    v8f acc0 = {}; v8f acc1 = {};
    for (int k0 = 0; k0 < DIN; k0 += 32) {
        const float* src = (k0 < G) ? XS : XT;
        const int kr = (k0 < G) ? k0 : (k0 - G);
        // A tile: 32 k-rows x 64 m  (16 chunks of 4 floats per row)
#pragma unroll
        for (int p = 0; p < 2; ++p) {
            int e = tid + p * 256;
            int k = e >> 4, mc = (e & 15) * 4;
            async_copy16(&src[(size_t)(kr + k) * B + m0 + mc], &sA[k][mc]);
        }
        // B tile: 32 k-rows x 64 n
#pragma unroll
        for (int p = 0; p < 2; ++p) {
            int e = tid + p * 256;
            int k = e >> 4, nc = (e & 15) * 4;
            async_copy16(&W[(size_t)(k0 + k) * H1N + n0 + nc], &sB[k][nc]);
        }
        async_wait();
        __syncthreads();
        const int ks = (lane >> 4) << 1;
        const int mr = mw + (lane & 15);
        const int nr = nw + (lane & 15);
#pragma unroll
        for (int kk = 0; kk < 32; kk += 4) {
            v2f a, b0, b1;
            a.x  = sA[kk + ks][mr];      a.y  = sA[kk + ks + 1][mr];
            b0.x = sB[kk + ks][nr];      b0.y = sB[kk + ks + 1][nr];
            b1.x = sB[kk + ks][nr + 16]; b1.y = sB[kk + ks + 1][nr + 16];
            acc0 = __builtin_amdgcn_wmma_f32_16x16x4_f32(false, a, false, b0, (short)0, acc0, false, false);
            acc1 = __builtin_amdgcn_wmma_f32_16x16x4_f32(false, a, false, b1, (short)0, acc1, false, false);
        }
        __syncthreads();
    }
    const int mbase = m0 + mw + ((lane & 16) ? 8 : 0);
    const int ncol  = n0 + nw + (lane & 15);
#pragma unroll
    for (int i = 0; i < 8; ++i) {
        out[(size_t)(mbase + i) * H1N + ncol]      = fmaxf(acc0[i] + bias[ncol], 0.f);
        out[(size_t)(mbase + i) * H1N + ncol + 16] = fmaxf(acc1[i] + bias[ncol + 16], 0.f);
    }
}

// ---------------- WMMA GEMM (row-major A): out = relu(A @ W + b) -------------------
// sA[m][k] stride 36 (16B-aligned chunks, conflict-free b64 frag loads),
// sB[k][n] stride 68.

__global__ void k_gemm_rm(const float* __restrict__ A, const float* __restrict__ W,
                          const float* __restrict__ bias, float* __restrict__ out,
                          int N, int Kdim) {
    __shared__ __align__(16) float sA[64][36];
    __shared__ __align__(16) float sB[32][68];
    const int tid  = threadIdx.x;
    const int lane = tid & 31;
    const int wv   = tid >> 5;
    const int m0 = blockIdx.x * 64;
    const int n0 = blockIdx.y * 64;
    const int mw = (wv & 3) * 16;
    const int nw = (wv >> 2) * 32;
    v8f acc0 = {}; v8f acc1 = {};
    for (int k0 = 0; k0 < Kdim; k0 += 32) {
        // A tile: 64 m-rows x 32 k (8 chunks per row)
#pragma unroll
        for (int p = 0; p < 2; ++p) {
            int e = tid + p * 256;
            int m = e >> 3, kc = (e & 7) * 4;
            async_copy16(&A[(size_t)(m0 + m) * Kdim + k0 + kc], &sA[m][kc]);
        }
        // B tile: 32 k-rows x 64 n
#pragma unroll
        for (int p = 0; p < 2; ++p) {
            int e = tid + p * 256;
            int k = e >> 4, nc = (e & 15) * 4;
            async_copy16(&W[(size_t)(k0 + k) * N + n0 + nc], &sB[k][nc]);
        }
        async_wait();
        __syncthreads();
        const int ks = (lane >> 4) << 1;
        const int mr = mw + (lane & 15);
        const int nr = nw + (lane & 15);
#pragma unroll
        for (int kk = 0; kk < 32; kk += 4) {
            v2f a = *(const v2f*)&sA[mr][kk + ks];
            v2f b0, b1;
            b0.x = sB[kk + ks][nr];      b0.y = sB[kk + ks + 1][nr];
            b1.x = sB[kk + ks][nr + 16]; b1.y = sB[kk + ks + 1][nr + 16];
            acc0 = __builtin_amdgcn_wmma_f32_16x16x4_f32(false, a, false, b0, (short)0, acc0, false, false);
            acc1 = __builtin_amdgcn_wmma_f32_16x16x4_f32(false, a, false, b1, (short)0, acc1, false, false);
        }
        __syncthreads();
    }
    const int mbase = m0 + mw + ((lane & 16) ? 8 : 0);
    const int ncol  = n0 + nw + (lane & 15);
#pragma unroll
    for (int i = 0; i < 8; ++i) {
        out[(size_t)(mbase + i) * N + ncol]      = fmaxf(acc0[i] + bias[ncol], 0.f);
        out[(size_t)(mbase + i) * N + ncol + 16] = fmaxf(acc1[i] + bias[ncol + 16], 0.f);
    }
}

// ---------------- tiny final layer: out[b,j] = h2[b,:] . w3[:,j] + b3[j] -----------

__global__ void k_gemm3(const float* __restrict__ Hm, const float* __restrict__ W,
                        const float* __restrict__ bias, float* __restrict__ out) {
    __shared__ float row[H2N];
    int b = blockIdx.x;
    for (int k = threadIdx.x; k < H2N; k += 64) row[k] = Hm[(size_t)b * H2N + k];
    __syncthreads();
    int j = threadIdx.x;
    if (j < NCLS) {
        float acc = bias[j];
        for (int k = 0; k < H2N; ++k) acc += row[k] * W[k * NCLS + j];
        out[(size_t)b * NCLS + j] = acc;
    }
}

// ---------------- launch ------------------------------------------------------------

extern "C" void kernel_launch(void* const* d_in, const int* in_sizes, int n_in,
                              void* d_out, int out_size, void* d_ws, size_t ws_size,
                              hipStream_t stream) {
    const float* x_sample = (const float*)d_in[0];
    const float* x_TF     = (const float*)d_in[1];
    const int*   adj_rows = (const int*)d_in[2];
    const int*   adj_cols = (const int*)d_in[3];
    const float* adj_vals = (const float*)d_in[4];
    const float* emb_mut  = (const float*)d_in[5];
    const float* bias_mut = (const float*)d_in[6];
    const float* emb_exp  = (const float*)d_in[7];
    const float* bias_exp = (const float*)d_in[8];
    const float* w1 = (const float*)d_in[9];
    const float* b1 = (const float*)d_in[10];
    const float* w2 = (const float*)d_in[11];
    const float* b2 = (const float*)d_in[12];
    const float* w3 = (const float*)d_in[13];
    const float* b3 = (const float*)d_in[14];
    float* out = (float*)d_out;

    char* wp = (char*)d_ws;
    auto carve = [&](size_t bytes) -> void* {
        void* p = (void*)wp;
        wp += (bytes + 255) & ~(size_t)255;
        return p;
    };
    const size_t GB = (size_t)G * B;
    float* xsA = (float*)carve(GB * 4);
    float* xsB = (float*)carve(GB * 4);
    float* xtA = (float*)carve(GB * 4);
    float* xtB = (float*)carve(GB * 4);
    float* h1  = (float*)carve((size_t)B * H1N * 4);
    float* h2  = (float*)carve((size_t)B * H2N * 4);
    int*   tcols   = (int*)carve((size_t)NNZ * 4);
    float* tvals   = (float*)carve((size_t)NNZ * 4);
    int*   tidx    = (int*)carve((size_t)NNZ * 4);
    int*   fwd_ptr = (int*)carve((size_t)(G + 1) * 4);
    int*   tr_ptr  = (int*)carve((size_t)(G + 1) * 4);
    int*   cursor  = (int*)carve((size_t)G * 4);

    // ---- build CSR for A (fwd) and A^T (deterministic) ----
    k_zero_ptrs<<<(G + 256) / 256, 256, 0, stream>>>(fwd_ptr, tr_ptr);
    k_count<<<(NNZ + 255) / 256, 256, 0, stream>>>(adj_rows, adj_cols, fwd_ptr, tr_ptr);
    k_scan2<<<2, 256, 0, stream>>>(fwd_ptr, tr_ptr);
    k_init_cursor<<<(G + 255) / 256, 256, 0, stream>>>(tr_ptr, cursor);
    k_scatter_tr<<<(NNZ + 255) / 256, 256, 0, stream>>>(adj_rows, adj_cols, adj_vals,
                                                        cursor, tcols, tvals, tidx);
    k_sort_seg<<<(G + 255) / 256, 256, 0, stream>>>(tr_ptr, tcols, tvals, tidx);

    // ---- embed + transpose to [G,B] ----
    dim3 tgrid(G / 32, B / 32), tblk(32, 8);
    k_embed_t<<<tgrid, tblk, 0, stream>>>(x_sample, emb_mut, bias_mut, xsA);
    k_embed_t<<<tgrid, tblk, 0, stream>>>(x_TF, emb_exp, bias_exp, xtA);

    // ---- K propagation steps ----
    float* xs_cur = xsA; float* xs_nxt = xsB;
    float* xt_cur = xtA; float* xt_nxt = xtB;
    dim3 sgrid(G, B / 1024);   // 4 float4-groups of 256 threads cover 4096 cols
    for (int it = 0; it < KSTEPS; ++it) {
        k_spmm<<<sgrid, 256, 0, stream>>>(fwd_ptr, adj_cols, adj_vals, xs_cur, xs_nxt);
        k_spmm<<<sgrid, 256, 0, stream>>>(tr_ptr, tcols, tvals, xt_cur, xt_nxt);
        float* t;
        t = xs_cur; xs_cur = xs_nxt; xs_nxt = t;
        t = xt_cur; xt_cur = xt_nxt; xt_nxt = t;
    }

    // ---- residual: x += alpha * embed(x_in) ----
    k_axpy_embed<<<tgrid, tblk, 0, stream>>>(x_sample, emb_mut, bias_mut, xs_cur);
    k_axpy_embed<<<tgrid, tblk, 0, stream>>>(x_TF, emb_exp, bias_exp, xt_cur);

    // ---- MLP ----
    k_gemm1<<<dim3(B / 64, H1N / 64), 256, 0, stream>>>(xs_cur, xt_cur, w1, b1, h1);
    k_gemm_rm<<<dim3(B / 64, H2N / 64), 256, 0, stream>>>(h1, w2, b2, h2, H2N, H1N);
    k_gemm3<<<B, 64, 0, stream>>>(h2, w3, b3, out);
}